// GCNEncoder_75728863363600
// MI455X (gfx1250) — compile-verified
//
#include <hip/hip_runtime.h>
#include <stddef.h>

typedef float v2f __attribute__((ext_vector_type(2)));
typedef float v8f __attribute__((ext_vector_type(8)));

// ---------------------------------------------------------------------------
// Utility: zero a float buffer
// ---------------------------------------------------------------------------
__global__ void zero_f32(float* __restrict__ p, int n) {
    int i = blockIdx.x * blockDim.x + threadIdx.x;
    if (i < n) p[i] = 0.0f;
}

// ---------------------------------------------------------------------------
// Degree count (after self-loop handled analytically later): deg[dst] += 1
// ---------------------------------------------------------------------------
__global__ void degree_k(const int* __restrict__ dst, int E, float* __restrict__ deg) {
    int e = blockIdx.x * blockDim.x + threadIdx.x;
    if (e < E) atomicAdd(&deg[dst[e]], 1.0f);
}

// dinv[i] = rsqrt(deg[i] + 1)   (in-place over deg buffer)
__global__ void rsqrt_k(float* __restrict__ d, int n) {
    int i = blockIdx.x * blockDim.x + threadIdx.x;
    if (i < n) d[i] = rsqrtf(d[i] + 1.0f);
}

// ---------------------------------------------------------------------------
// Dense GEMM H[M,N] = X[M,K] * W[K,N] using V_WMMA_F32_16X16X4_F32.
// One wave32 computes one 16x16 output tile. Requires M%16==0, N%16==0, K%4==0.
// A layout (16x4 f32): lane l<16 holds K={k,k+1}, lane l>=16 holds K={k+2,k+3}
// for row M = l&15.  B layout mirrors: lane = column, halves split K.
// C/D layout: VGPR v -> row M = v + 8*(lane>>4), col N = lane&15.
// ---------------------------------------------------------------------------
__global__ __launch_bounds__(256) void gemm_wmma_f32(
    const float* __restrict__ X, const float* __restrict__ W,
    float* __restrict__ H, int M, int K, int N) {
    int gwave = (int)((blockIdx.x * (unsigned)blockDim.x + threadIdx.x) >> 5);
    int lane  = threadIdx.x & 31;
    int tilesN = N >> 4;
    int tilesM = M >> 4;
    if (gwave >= tilesM * tilesN) return;   // uniform per wave: EXEC stays all-1s
    int tm = gwave / tilesN;
    int tn = gwave - tm * tilesN;
    int row0 = tm << 4, col0 = tn << 4;
    int half = lane >> 4;     // 0: K pair {0,1}, 1: K pair {2,3}
    int l    = lane & 15;

    const float* xrow = X + (size_t)(row0 + l) * K + 2 * half;   // A row for this lane
    const float* wcol = W + col0 + l;                             // B column for this lane

    v8f acc = {};
    for (int k = 0; k < K; k += 4) {
        v2f a, b;
        a.x = xrow[k];
        a.y = xrow[k + 1];
        int kb = k + 2 * half;
        b.x = wcol[(size_t)kb * N];
        b.y = wcol[(size_t)(kb + 1) * N];
        acc = __builtin_amdgcn_wmma_f32_16x16x4_f32(
            /*neg_a=*/false, a, /*neg_b=*/false, b,
            /*c_mod=*/(short)0, acc, /*reuse_a=*/false, /*reuse_b=*/false);
    }

    float* hp = H + (size_t)(row0 + 8 * half) * N + col0 + l;
#pragma unroll
    for (int v = 0; v < 8; ++v)
        hp[(size_t)v * N] = acc[v];
}

// ---------------------------------------------------------------------------
// Edge scatter: agg[dst] += h[src] * dinv[src] * dinv[dst]
// One thread per (edge, 4-feature chunk). F and chunk count are powers of two.
// Feature matrices (<= 51 MB) are L2-resident on MI455X (192 MB L2), so both
// the random gather and the float atomics resolve in L2.
// ---------------------------------------------------------------------------
__global__ void scatter_k(const float* __restrict__ h, const int* __restrict__ src,
                          const int* __restrict__ dst, const float* __restrict__ dinv,
                          float* __restrict__ agg, int E, int F, int log2Chunks) {
    int tid = blockIdx.x * blockDim.x + threadIdx.x;
    int e = tid >> log2Chunks;
    if (e >= E) return;
    int c = (tid & ((1 << log2Chunks) - 1)) << 2;
    int s = src[e], d = dst[e];
    float nrm = dinv[s] * dinv[d];
    const float4 v = *(const float4*)(h + (size_t)s * F + c);
    float* ap = agg + (size_t)d * F + c;
    atomicAdd(ap + 0, v.x * nrm);
    atomicAdd(ap + 1, v.y * nrm);
    atomicAdd(ap + 2, v.z * nrm);
    atomicAdd(ap + 3, v.w * nrm);
}

// ---------------------------------------------------------------------------
// Fused epilogue: out = agg + h * dinv^2 + bias, optional ReLU.
// out may alias agg (in-place).
// ---------------------------------------------------------------------------
__global__ void finalize_k(float* __restrict__ out, const float* __restrict__ agg,
                           const float* __restrict__ h, const float* __restrict__ dinv,
                           const float* __restrict__ bias,
                           int total, int log2F, int doRelu) {
    int tid = blockIdx.x * blockDim.x + threadIdx.x;
    if (tid >= total) return;
    int node = tid >> log2F;
    int f = tid & ((1 << log2F) - 1);
    float di = dinv[node];
    float v = agg[tid] + h[tid] * di * di + bias[f];
    if (doRelu) v = fmaxf(v, 0.0f);
    out[tid] = v;
}

// ---------------------------------------------------------------------------
// Host-side orchestration
// ---------------------------------------------------------------------------
static inline int cdiv(long long a, int b) { return (int)((a + b - 1) / b); }

extern "C" void kernel_launch(void* const* d_in, const int* in_sizes, int n_in,
                              void* d_out, int out_size, void* d_ws, size_t ws_size,
                              hipStream_t stream) {
    const float* x   = (const float*)d_in[0];
    const int*   ei  = (const int*)d_in[1];
    const float* W1  = (const float*)d_in[2];
    const float* b1  = (const float*)d_in[3];
    const float* W2  = (const float*)d_in[4];
    const float* b2  = (const float*)d_in[5];
    const float* W3  = (const float*)d_in[6];
    const float* b3  = (const float*)d_in[7];
    float* out = (float*)d_out;

    const int IN = 128, H1 = 64, H2 = 128, OUT = 32;
    const int Nn = in_sizes[0] / IN;      // 100000
    const int E  = in_sizes[1] / 2;       // 1,600,000
    const int* src = ei;
    const int* dst = ei + E;

    // Workspace layout (floats): dinv[Nn] | P[Nn*128] | Q[Nn*128]
    float* dinv = (float*)d_ws;
    float* P = dinv + Nn;
    float* Q = P + (size_t)Nn * 128;

    const int B = 256;

    // --- degrees -> dinv ---
    zero_f32<<<cdiv(Nn, B), B, 0, stream>>>(dinv, Nn);
    degree_k<<<cdiv(E, B), B, 0, stream>>>(dst, E, dinv);
    rsqrt_k<<<cdiv(Nn, B), B, 0, stream>>>(dinv, Nn);

    // --- Layer 1: 128 -> 64, ReLU ---
    {
        int tiles = (Nn >> 4) * (H1 >> 4);
        gemm_wmma_f32<<<cdiv((long long)tiles * 32, B), B, 0, stream>>>(x, W1, P, Nn, IN, H1);
        zero_f32<<<cdiv((long long)Nn * H1, B), B, 0, stream>>>(Q, Nn * H1);
        scatter_k<<<cdiv((long long)E * (H1 >> 2), B), B, 0, stream>>>(P, src, dst, dinv, Q, E, H1, 4);
        finalize_k<<<cdiv((long long)Nn * H1, B), B, 0, stream>>>(Q, Q, P, dinv, b1, Nn * H1, 6, 1);
    }

    // --- Layer 2: 64 -> 128, ReLU ---
    {
        int tiles = (Nn >> 4) * (H2 >> 4);
        gemm_wmma_f32<<<cdiv((long long)tiles * 32, B), B, 0, stream>>>(Q, W2, P, Nn, H1, H2);
        zero_f32<<<cdiv((long long)Nn * H2, B), B, 0, stream>>>(Q, Nn * H2);
        // NOTE: GEMM read of Q completes before zero (stream order), so reuse is safe.
        scatter_k<<<cdiv((long long)E * (H2 >> 2), B), B, 0, stream>>>(P, src, dst, dinv, Q, E, H2, 5);
        finalize_k<<<cdiv((long long)Nn * H2, B), B, 0, stream>>>(Q, Q, P, dinv, b2, Nn * H2, 7, 1);
    }

    // --- Layer 3: 128 -> 32, no ReLU; aggregate straight into d_out ---
    {
        int tiles = (Nn >> 4) * (OUT >> 4);
        gemm_wmma_f32<<<cdiv((long long)tiles * 32, B), B, 0, stream>>>(Q, W3, P, Nn, H2, OUT);
        zero_f32<<<cdiv((long long)Nn * OUT, B), B, 0, stream>>>(out, Nn * OUT);
        scatter_k<<<cdiv((long long)E * (OUT >> 2), B), B, 0, stream>>>(P, src, dst, dinv, out, E, OUT, 3);
        finalize_k<<<cdiv((long long)Nn * OUT, B), B, 0, stream>>>(out, out, P, dinv, b3, Nn * OUT, 5, 0);
    }
}